// SE3Comp_79242146611602
// MI455X (gfx1250) — compile-verified
//
#include <hip/hip_runtime.h>

typedef __attribute__((ext_vector_type(2))) float v2f;
typedef __attribute__((ext_vector_type(8))) float v8f;
typedef __attribute__((ext_vector_type(4))) int   v4i;

#define SE3_EPS 1e-12f

// CDNA5 async global<->LDS paths (ASYNCcnt), with synchronous fallback.
#if __has_builtin(__builtin_amdgcn_global_load_async_to_lds_b32) && \
    __has_builtin(__builtin_amdgcn_s_wait_asynccnt)
#define HAVE_ASYNC_LOAD 1
#else
#define HAVE_ASYNC_LOAD 0
#endif
#if __has_builtin(__builtin_amdgcn_global_store_async_from_lds_b128) && \
    __has_builtin(__builtin_amdgcn_s_wait_asynccnt)
#define HAVE_ASYNC_STORE 1
#else
#define HAVE_ASYNC_STORE 0
#endif

// Typed address-space pointers matching the builtin signatures
// (load_b32: as1 int* src; store_b128: as1 int4* dst).
typedef __attribute__((address_space(1))) int as1_int;
typedef __attribute__((address_space(3))) int as3_int;
typedef __attribute__((address_space(1))) v4i as1_v4i;
typedef __attribute__((address_space(3))) v4i as3_v4i;

// Block = 256 threads = 8 wave32. Each block processes 256 batch elements.
// Phase 0: stage inputs into LDS (async DMA when available, coalesced).
// Phase 1: per-lane SE(3) exp + quaternion matrix (VALU).
// Phase 2: batched 4x4 compose via V_WMMA_F32_16X16X4_F32 (4 elements/WMMA,
//          diagonal 4x4 blocks of the 16x16 product are the results).
// Phase 3: coalesced 16B copy-out of the LDS result tile (async when available).
__global__ __launch_bounds__(256) void se3_compose_wmma(
    const float* __restrict__ Tg, const float* __restrict__ xi,
    float* __restrict__ out, int Bn)
{
    __shared__ float sTg [256 * 7];        //  7168 B  staged Tg rows
    __shared__ float sXi [256 * 6];        //  6144 B  staged xi rows
    __shared__ float sTxi[256 * 13 + 8];   // Txi rows 0..2, stride 13 (+pad for safe loads)
    __shared__ float sTgm[256 * 13 + 8];   // Tgm rows 0..2, stride 13 (+pad)
    __shared__ __align__(16) float sRes[256 * 16];   // result tile, row-major per element

    const int       tid  = threadIdx.x;
    const long long base = (long long)blockIdx.x * 256;
    const bool      full = (base + 256) <= (long long)Bn;   // block-uniform

    // ---------------- phase 0: stage inputs ----------------
    {
        const float* __restrict__ tgp = Tg + base * 7;
        const float* __restrict__ xip = xi + base * 6;
        if (full) {
#if HAVE_ASYNC_LOAD
            #pragma unroll
            for (int i = tid; i < 256 * 7; i += 256)
                __builtin_amdgcn_global_load_async_to_lds_b32(
                    (as1_int*)(tgp + i), (as3_int*)(&sTg[i]), 0, 0);
            #pragma unroll
            for (int i = tid; i < 256 * 6; i += 256)
                __builtin_amdgcn_global_load_async_to_lds_b32(
                    (as1_int*)(xip + i), (as3_int*)(&sXi[i]), 0, 0);
            __builtin_amdgcn_s_wait_asynccnt(0);
#else
            #pragma unroll
            for (int i = tid; i < 256 * 7; i += 256) sTg[i] = tgp[i];
            #pragma unroll
            for (int i = tid; i < 256 * 6; i += 256) sXi[i] = xip[i];
#endif
        } else {
            const int remT = (int)((long long)Bn * 7 - base * 7);
            const int remX = (int)((long long)Bn * 6 - base * 6);
            for (int i = tid; i < 256 * 7; i += 256) sTg[i] = (i < remT) ? tgp[i] : 0.f;
            for (int i = tid; i < 256 * 6; i += 256) sXi[i] = (i < remX) ? xip[i] : 0.f;
        }
    }
    __syncthreads();

    // ---------------- phase 1: per-lane matrix construction ----------------
    {
        const float rx = sXi[tid * 6 + 0], ry = sXi[tid * 6 + 1], rz = sXi[tid * 6 + 2];
        const float wx = sXi[tid * 6 + 3], wy = sXi[tid * 6 + 4], wz = sXi[tid * 6 + 5];
        const float tx = sTg[tid * 7 + 0], ty = sTg[tid * 7 + 1], tz = sTg[tid * 7 + 2];
        const float qw = sTg[tid * 7 + 3], qx = sTg[tid * 7 + 4];
        const float qy = sTg[tid * 7 + 5], qz = sTg[tid * 7 + 6];

        const float th2  = wx * wx + wy * wy + wz * wz;
        const bool  safe = th2 > SE3_EPS;
        const float ts   = safe ? th2 : 1.f;
        const float t    = sqrtf(ts);
        const float sn   = __sinf(t);
        const float cs   = __cosf(t);
        const float Af   = safe ? (sn / t) : 1.f;
        const float Bf   = safe ? ((1.f - cs) / ts) : 0.f;
        const float Cf   = safe ? ((t - sn) / (ts * t)) : (1.f / 6.f);

        // K^2 = w w^T - th2 * I
        const float xx = wx * wx - th2, yy = wy * wy - th2, zz = wz * wz - th2;
        const float xy = wx * wy, xz = wx * wz, yz = wy * wz;

        // R = I + A*K + B*K^2 ; V = I + B*K + C*K^2 ; t_xi = V * rho
        const float R00 = 1.f + Bf * xx;      const float R01 = -Af * wz + Bf * xy;
        const float R02 =  Af * wy + Bf * xz; const float R10 =  Af * wz + Bf * xy;
        const float R11 = 1.f + Bf * yy;      const float R12 = -Af * wx + Bf * yz;
        const float R20 = -Af * wy + Bf * xz; const float R21 =  Af * wx + Bf * yz;
        const float R22 = 1.f + Bf * zz;

        const float V00 = 1.f + Cf * xx;      const float V01 = -Bf * wz + Cf * xy;
        const float V02 =  Bf * wy + Cf * xz; const float V10 =  Bf * wz + Cf * xy;
        const float V11 = 1.f + Cf * yy;      const float V12 = -Bf * wx + Cf * yz;
        const float V20 = -Bf * wy + Cf * xz; const float V21 =  Bf * wx + Cf * yz;
        const float V22 = 1.f + Cf * zz;

        const float t0 = V00 * rx + V01 * ry + V02 * rz;
        const float t1 = V10 * rx + V11 * ry + V12 * rz;
        const float t2 = V20 * rx + V21 * ry + V22 * rz;

        float* A = &sTxi[tid * 13];
        A[0] = R00; A[1] = R01; A[2]  = R02; A[3]  = t0;
        A[4] = R10; A[5] = R11; A[6]  = R12; A[7]  = t1;
        A[8] = R20; A[9] = R21; A[10] = R22; A[11] = t2;

        float* Bm = &sTgm[tid * 13];
        Bm[0]  = 1.f - 2.f * (qy * qy + qz * qz);
        Bm[1]  = 2.f * (qx * qy - qz * qw);
        Bm[2]  = 2.f * (qx * qz + qy * qw);
        Bm[3]  = tx;
        Bm[4]  = 2.f * (qx * qy + qz * qw);
        Bm[5]  = 1.f - 2.f * (qx * qx + qz * qz);
        Bm[6]  = 2.f * (qy * qz - qx * qw);
        Bm[7]  = ty;
        Bm[8]  = 2.f * (qx * qz - qy * qw);
        Bm[9]  = 2.f * (qy * qz + qx * qw);
        Bm[10] = 1.f - 2.f * (qx * qx + qy * qy);
        Bm[11] = tz;
    }
    __syncthreads();

    // ---------------- phase 2: WMMA batched 4x4 composes ----------------
    // A (16x4 f32, 2 VGPRs): lane L holds A[m][k], m = L%16, k = vgpr + 2*(L/16)
    // B (4x16 f32, 2 VGPRs): lane L holds B[k][n], n = L%16, k = vgpr + 2*(L/16)
    // D (16x16 f32, 8 VGPRs): lane L holds D[m][n], n = L%16, m = vgpr + 8*(L/16)
    {
        const int wv  = tid >> 5;        // wave 0..7
        const int ln  = tid & 31;
        const int hi  = ln >> 4;         // lane half
        const int q16 = ln & 15;
        const int blk = q16 >> 2;        // 4x4 block id 0..3
        const int rc  = q16 & 3;         // A row within block == B col within block
        const int k0  = 2 * hi;          // k of vgpr0; vgpr1 -> k0+1

        const bool  useful  = hi ? (blk >= 2) : (blk < 2);
        const bool  lowhalf = (blk & 1) == 0;           // D rows in d[0..3] vs d[4..7]
        const bool  row3    = (rc == 3);                // synthesized Txi bottom row
        const float a1c     = hi ? 1.f : 0.f;           // A[3][k0+1]: 1 iff k0+1==3
        const float b1c     = (rc == 3) ? 1.f : 0.f;    // B[3][n]:   1 iff col==3
        const bool  kIs3    = (hi != 0);                // k0+1 == 3

        int fb = (wv * 32 + blk) * 13;   // fragment base, += 52 per group
        int eo = (wv * 32 + blk) * 16;   // result base,   += 64 per group
        const int aoff = rc * 4 + k0;
        const int boff = k0 * 4 + rc;

        #pragma unroll
        for (int g = 0; g < 8; ++g, fb += 52, eo += 64) {
            // unconditional LDS loads (padded arrays keep rc==3 / k==3 addrs in-bounds)
            const float a0 = sTxi[fb + aoff];
            const float a1 = sTxi[fb + aoff + 1];
            const float b0 = sTgm[fb + boff];
            const float b1 = sTgm[fb + boff + 4];

            v2f a, b;
            a.x = row3 ? 0.f : a0;
            a.y = row3 ? a1c : a1;
            b.x = b0;                    // k0 in {0,2}: always a stored row
            b.y = kIs3 ? b1c : b1;

            v8f acc = {0.f, 0.f, 0.f, 0.f, 0.f, 0.f, 0.f, 0.f};
            v8f d = __builtin_amdgcn_wmma_f32_16x16x4_f32(
                /*neg_a=*/false, a, /*neg_b=*/false, b,
                /*c_mod=*/(short)0, acc, /*reuse_a=*/false, /*reuse_b=*/false);

            // half-select with constant extract indices (v_cndmask, no dyn indexing)
            const float r0 = lowhalf ? d[0] : d[4];
            const float r1 = lowhalf ? d[1] : d[5];
            const float r2 = lowhalf ? d[2] : d[6];
            const float r3 = lowhalf ? d[3] : d[7];

            if (useful) {
                float* Rr = &sRes[eo];
                Rr[rc]      = r0;
                Rr[4 + rc]  = r1;
                Rr[8 + rc]  = r2;
                Rr[12 + rc] = r3;
            }
        }
    }
    __syncthreads();

    // ---------------- phase 3: coalesced 16B copy-out ----------------
    {
        float* __restrict__ op = out + base * 16;
        if (full) {
#if HAVE_ASYNC_STORE
            #pragma unroll
            for (int q = 0; q < 4; ++q) {
                const int i = (q * 256 + tid) * 4;   // float index, 16B aligned
                __builtin_amdgcn_global_store_async_from_lds_b128(
                    (as1_v4i*)(op + i), (as3_v4i*)(&sRes[i]), 0, 0);
            }
            __builtin_amdgcn_s_wait_asynccnt(0);
#else
            #pragma unroll
            for (int q = 0; q < 4; ++q) {
                const int i4 = q * 256 + tid;
                ((float4*)op)[i4] = ((const float4*)sRes)[i4];
            }
#endif
        } else {
            const int remE = (int)((long long)Bn - base);   // elements in this block
            #pragma unroll
            for (int q = 0; q < 4; ++q) {
                const int i4 = q * 256 + tid;               // float4 index
                if ((i4 >> 2) < remE) ((float4*)op)[i4] = ((const float4*)sRes)[i4];
            }
        }
    }
}

extern "C" void kernel_launch(void* const* d_in, const int* in_sizes, int n_in,
                              void* d_out, int out_size, void* d_ws, size_t ws_size,
                              hipStream_t stream) {
    (void)n_in; (void)d_ws; (void)ws_size; (void)out_size;
    const float* Tg = (const float*)d_in[0];   // (B,7) f32
    const float* xi = (const float*)d_in[1];   // (B,6) f32
    float* out = (float*)d_out;                // (B,4,4) f32
    const int B = in_sizes[0] / 7;
    const int blocks = (B + 255) / 256;
    se3_compose_wmma<<<blocks, 256, 0, stream>>>(Tg, xi, out, B);
}